// LocalInteraction_49160195670227
// MI455X (gfx1250) — compile-verified
//
#include <hip/hip_runtime.h>
#include <hip/hip_bf16.h>

// ---------------------------------------------------------------------------
// CDNA5 / gfx1250 implementation.
//  - All GEMMs on V_WMMA_F32_16X16X4_F32 (fp32-exact matrix path).
//  - Weight/parameter staging via GLOBAL_LOAD_ASYNC_TO_LDS_B128 (ASYNCcnt),
//    overlapped with VALU activation work, then s_wait_asynccnt + barrier.
// ---------------------------------------------------------------------------

typedef __attribute__((ext_vector_type(2))) float v2f;
typedef __attribute__((ext_vector_type(8))) float v8f;

#define DIM   128
#define LDP   132          // padded LDS row stride (floats) -> conflict-free strided reads
#define NATOMS 16384
#define NEDGES 100000

__constant__ float BINOM15[16] = {1.f,15.f,105.f,455.f,1365.f,3003.f,5005.f,6435.f,
                                  6435.f,5005.f,3003.f,1365.f,455.f,105.f,15.f,1.f};

__device__ __forceinline__ float swishf(float x, float a, float b) {
    return a * x * (1.0f / (1.0f + __expf(-b * x)));
}

// ---- CDNA5 async memory->LDS DMA helpers (no VGPR data path, ASYNCcnt) ----
__device__ __forceinline__ void async_b128_to_lds(uint32_t lds_off, const float* gptr) {
    asm volatile("global_load_async_to_lds_b128 %0, %1, off"
                 :: "v"(lds_off), "v"(gptr) : "memory");
}
__device__ __forceinline__ void wait_async0() {
    asm volatile("s_wait_asynccnt 0x0" ::: "memory");
}
// Copy a 128x128 f32 matrix (row-major, contiguous) into LDS with row stride
// LDP floats, using 16B async chunks. 4096 chunks over 256 threads.
__device__ __forceinline__ void async_copy_mat128(float* lds_base, const float* g, int tid) {
    const uint32_t base = (uint32_t)(uintptr_t)lds_base;   // low 32b of LDS aperture = offset
#pragma unroll
    for (int it = 0; it < 16; ++it) {
        const int c = tid + it * 256;          // 0..4095
        const int r = c >> 5, q = c & 31;      // row, 16B-chunk within row
        async_b128_to_lds(base + (uint32_t)(r * LDP + q * 4) * 4u,
                          g + (size_t)r * DIM + q * 4);
    }
}

// C(16x16) accumulate over K=128: both operands row-major with a K index,
// A[m][k] (lda) and B[n][k] (ldb).  WMMA f32 16x16x4 per-lane layout:
//   lanes 0-15: m/n=lane, k-pair {4kk,4kk+1}; lanes 16-31: m/n=lane-16, {4kk+2,4kk+3}
__device__ __forceinline__ v8f wmma_krow(const float* __restrict__ A, int lda,
                                         const float* __restrict__ B, int ldb,
                                         v8f acc, int lane) {
    const int hi = lane >> 4, mr = lane & 15, kb = 2 * hi;
    const float* ar = A + mr * lda;
    const float* br = B + mr * ldb;
#pragma unroll
    for (int kk = 0; kk < 32; ++kk) {
        const int k = 4 * kk + kb;
        v2f av; av.x = ar[k]; av.y = ar[k + 1];
        v2f bv; bv.x = br[k]; bv.y = br[k + 1];
        acc = __builtin_amdgcn_wmma_f32_16x16x4_f32(false, av, false, bv,
                                                    (short)0, acc, false, false);
    }
    return acc;
}

// Same but B stored K-major: B[k][n] with leading dim ldb (pre-offset to column base).
__device__ __forceinline__ v8f wmma_kcol(const float* __restrict__ A, int lda,
                                         const float* __restrict__ B, int ldb,
                                         v8f acc, int lane) {
    const int hi = lane >> 4, mr = lane & 15, kb = 2 * hi;
    const float* ar = A + mr * lda;
    const float* bc = B + mr;           // column n = mr
#pragma unroll
    for (int kk = 0; kk < 32; ++kk) {
        const int k = 4 * kk + kb;
        v2f av; av.x = ar[k]; av.y = ar[k + 1];
        v2f bv; bv.x = bc[k * ldb]; bv.y = bc[(k + 1) * ldb];
        acc = __builtin_amdgcn_wmma_f32_16x16x4_f32(false, av, false, bv,
                                                    (short)0, acc, false, false);
    }
    return acc;
}

// ---------------------------------------------------------------------------
// Residual MLP: Y = swish2(X + swish1(swish0(X)@W1^T + b1)@W2^T + b2) @ Wout^T
// 64 rows per workgroup, 8 waves = 8 column tiles. Weights DMA'd to LDS
// asynchronously while the activation VALU work runs.
// ---------------------------------------------------------------------------
__global__ __launch_bounds__(256) void resmlp_kernel(
    const float* __restrict__ X,
    const float* __restrict__ W1, const float* __restrict__ b1,
    const float* __restrict__ W2, const float* __restrict__ b2,
    const float* __restrict__ Wout,
    const float* __restrict__ act_a, const float* __restrict__ act_b,
    float* __restrict__ Y) {
    __shared__ float Xact[64 * LDP];
    __shared__ float Hs[64 * LDP];
    __shared__ float Ws[DIM * LDP];

    const int tid = threadIdx.x;
    const int lane = tid & 31;
    const int wave = tid >> 5;            // column tile 0..7
    const int row0 = blockIdx.x * 64;

    const float a0 = act_a[0], a1 = act_a[1], a2 = act_a[2];
    const float q0 = act_b[0], q1 = act_b[1], q2 = act_b[2];
    const int hi = lane >> 4, mr = lane & 15;

    // stage 0: DMA W1 -> Ws while computing Xact = swish0(X)
    async_copy_mat128(Ws, W1, tid);
    for (int i = tid; i < 64 * DIM; i += 256) {
        int rr = i >> 7, cc = i & 127;
        Xact[rr * LDP + cc] = swishf(X[(size_t)(row0 + rr) * DIM + cc], a0, q0);
    }
    wait_async0();
    __syncthreads();

    // GEMM 1: Hs = Xact @ W1^T + b1
    for (int rb = 0; rb < 4; ++rb) {
        v8f acc = {};
        acc = wmma_krow(&Xact[(rb * 16) * LDP], LDP, &Ws[(wave * 16) * LDP], LDP, acc, lane);
        float bias = b1[wave * 16 + mr];
#pragma unroll
        for (int vr = 0; vr < 8; ++vr)
            Hs[(rb * 16 + vr + 8 * hi) * LDP + wave * 16 + mr] = acc[vr] + bias;
    }
    __syncthreads();

    // stage 1: DMA W2 -> Ws while computing Xact = swish1(Hs)
    async_copy_mat128(Ws, W2, tid);
    for (int i = tid; i < 64 * DIM; i += 256) {
        int rr = i >> 7, cc = i & 127;
        Xact[rr * LDP + cc] = swishf(Hs[rr * LDP + cc], a1, q1);
    }
    wait_async0();
    __syncthreads();

    // GEMM 2: Hs = Xact @ W2^T + b2
    for (int rb = 0; rb < 4; ++rb) {
        v8f acc = {};
        acc = wmma_krow(&Xact[(rb * 16) * LDP], LDP, &Ws[(wave * 16) * LDP], LDP, acc, lane);
        float bias = b2[wave * 16 + mr];
#pragma unroll
        for (int vr = 0; vr < 8; ++vr)
            Hs[(rb * 16 + vr + 8 * hi) * LDP + wave * 16 + mr] = acc[vr] + bias;
    }
    __syncthreads();

    // stage 2: DMA Wout -> Ws while computing y = X + Hs ; Xact = swish2(y)
    async_copy_mat128(Ws, Wout, tid);
    for (int i = tid; i < 64 * DIM; i += 256) {
        int rr = i >> 7, cc = i & 127;
        float yv = X[(size_t)(row0 + rr) * DIM + cc] + Hs[rr * LDP + cc];
        Xact[rr * LDP + cc] = swishf(yv, a2, q2);
    }
    wait_async0();
    __syncthreads();

    // GEMM 3: Y = Xact @ Wout^T (straight to global)
    for (int rb = 0; rb < 4; ++rb) {
        v8f acc = {};
        acc = wmma_krow(&Xact[(rb * 16) * LDP], LDP, &Ws[(wave * 16) * LDP], LDP, acc, lane);
#pragma unroll
        for (int vr = 0; vr < 8; ++vr)
            Y[(size_t)(row0 + rb * 16 + vr + 8 * hi) * DIM + wave * 16 + mr] = acc[vr];
    }
}

// ---------------------------------------------------------------------------
// Edge kernel: one wave per edge (grid-stride). Computes rho (Bernstein x fcut),
// spherical harmonics, Grho matvecs from LDS, scatters f*Grho*y into quants
// (layout [atom][9 channels][128]) with global fp32 atomics (L2 resident).
// ---------------------------------------------------------------------------
__global__ __launch_bounds__(256) void edge_kernel(
    const float* __restrict__ xyz, const int* __restrict__ nbrs,
    const float* __restrict__ feat1, const float* __restrict__ feat2,
    const float* __restrict__ feat3,
    const float* __restrict__ G_s, const float* __restrict__ G_p,
    const float* __restrict__ G_d,
    const float* __restrict__ gamma_p,
    float* __restrict__ quants, int nedges) {
    __shared__ float Gsl[DIM * 17];
    __shared__ float Gpl[DIM * 17];
    __shared__ float Gdl[DIM * 17];
    const int tid = threadIdx.x;
    for (int i = tid; i < DIM * 16; i += 256) {
        int d = i >> 4, k = i & 15;
        Gsl[d * 17 + k] = G_s[i];
        Gpl[d * 17 + k] = G_p[i];
        Gdl[d * 17 + k] = G_d[i];
    }
    __syncthreads();

    const float gamma = gamma_p[0];
    const int lane = tid & 31;
    const int gwave = blockIdx.x * 8 + (tid >> 5);
    const int nwaves = gridDim.x * 8;

    for (int e = gwave; e < nedges; e += nwaves) {
        const int src = nbrs[2 * e];
        const int dst = nbrs[2 * e + 1];
        const float dx = xyz[3 * dst + 0] - xyz[3 * src + 0];
        const float dy = xyz[3 * dst + 1] - xyz[3 * src + 1];
        const float dz = xyz[3 * dst + 2] - xyz[3 * src + 2];
        const float r2 = dx * dx + dy * dy + dz * dz + 3.0f * 1e-15f;
        const float r = sqrtf(r2);
        const float inv = 1.0f / r;
        const float ux = dx * inv, uy = dy * inv, uz = dz * inv;

        // bernstein_rho
        const float x = __expf(-gamma * r);
        const float omx = 1.0f - x;
        float fcut = 0.0f;
        if (r < 5.0f) fcut = __expf(-r2 / (25.0f - r2));
        float rho[16];
        float xk = 1.0f;
#pragma unroll
        for (int k = 0; k < 16; ++k) { rho[k] = BINOM15[k] * xk; xk *= x; }
        float oj = fcut;                 // fold fcut into (1-x)^j accumulation
#pragma unroll
        for (int k = 15; k >= 0; --k) { rho[k] *= oj; oj *= omx; }

        const float s3 = 1.7320508075688772f;
        const float yp0 = uy, yp1 = uz, yp2 = ux;
        const float yd0 = s3 * ux * uy, yd1 = s3 * uy * uz;
        const float yd2 = 0.5f * (3.0f * uz * uz - 1.0f);
        const float yd3 = s3 * ux * uz, yd4 = 0.5f * s3 * (ux * ux - uy * uy);

        float* qb = quants + (size_t)src * (9 * DIM);
#pragma unroll
        for (int j = 0; j < 4; ++j) {
            const int d = lane + 32 * j;
            float gs = 0.f, gp = 0.f, gd = 0.f;
#pragma unroll
            for (int k = 0; k < 16; ++k) {
                const float rk = rho[k];
                gs += Gsl[d * 17 + k] * rk;
                gp += Gpl[d * 17 + k] * rk;
                gd += Gdl[d * 17 + k] * rk;
            }
            const float fs = feat1[(size_t)dst * DIM + d] * gs;
            const float fp = feat2[(size_t)dst * DIM + d] * gp;
            const float fd = feat3[(size_t)dst * DIM + d] * gd;
            atomicAdd(&qb[0 * DIM + d], fs);
            atomicAdd(&qb[1 * DIM + d], fp * yp0);
            atomicAdd(&qb[2 * DIM + d], fp * yp1);
            atomicAdd(&qb[3 * DIM + d], fp * yp2);
            atomicAdd(&qb[4 * DIM + d], fd * yd0);
            atomicAdd(&qb[5 * DIM + d], fd * yd1);
            atomicAdd(&qb[6 * DIM + d], fd * yd2);
            atomicAdd(&qb[7 * DIM + d], fd * yd3);
            atomicAdd(&qb[8 * DIM + d], fd * yd4);
        }
    }
}

// ---------------------------------------------------------------------------
// Inner-product kernel: 16 atoms per workgroup. Packs 16 atoms' q-matrices into
// B panels (p: 48 cols = 3 tiles, d: 80 cols = 5 tiles), GEMMs against P/D
// matrices (async-DMA'd into LDS), reduces t1*t2 over m with LDS fp32 atomics:
//   inp = c_term + q_s + inner_p + inner_d.
// ---------------------------------------------------------------------------
__global__ __launch_bounds__(256) void inner_kernel(
    const float* __restrict__ feat0, const float* __restrict__ quants,
    const float* __restrict__ P_1, const float* __restrict__ P_2,
    const float* __restrict__ D_1, const float* __restrict__ D_2,
    float* __restrict__ inp) {
    __shared__ float M1s[DIM * LDP];
    __shared__ float M2s[DIM * LDP];
    __shared__ float Bq[DIM * 80];
    __shared__ float accD[16 * 129];

    const int tid = threadIdx.x;
    const int lane = tid & 31;
    const int wave = tid >> 5;        // row tile 0..7
    const int hi = lane >> 4, mr = lane & 15;
    const int a0 = blockIdx.x * 16;

    // DMA P matrices while doing the gather/init VALU work
    async_copy_mat128(M1s, P_1, tid);
    async_copy_mat128(M2s, P_2, tid);
    // accD = c_term + q_s
    for (int i = tid; i < 16 * DIM; i += 256) {
        int t = i >> 7, d = i & 127;
        accD[t * 129 + d] = feat0[(size_t)(a0 + t) * DIM + d] +
                            quants[(size_t)(a0 + t) * (9 * DIM) + 0 * DIM + d];
    }
    // packed q_p panel (col = t*3 + m)
    for (int i = tid; i < DIM * 48; i += 256) {
        int k = i / 48, col = i - 48 * k;
        int t = col / 3, mm = col - 3 * t;
        Bq[k * 80 + col] = quants[(size_t)(a0 + t) * (9 * DIM) + (1 + mm) * DIM + k];
    }
    wait_async0();
    __syncthreads();

    // p part: 3 column tiles
    for (int ct = 0; ct < 3; ++ct) {
        v8f acc1 = {}, acc2 = {};
        acc1 = wmma_kcol(&M1s[(wave * 16) * LDP], LDP, &Bq[ct * 16], 80, acc1, lane);
        acc2 = wmma_kcol(&M2s[(wave * 16) * LDP], LDP, &Bq[ct * 16], 80, acc2, lane);
        const int col = ct * 16 + mr;
        const int t = col / 3;
#pragma unroll
        for (int vr = 0; vr < 8; ++vr)
            atomicAdd(&accD[t * 129 + wave * 16 + vr + 8 * hi], acc1[vr] * acc2[vr]);
    }
    __syncthreads();

    // DMA D matrices while rebuilding the q_d panel (col = t*5 + m)
    async_copy_mat128(M1s, D_1, tid);
    async_copy_mat128(M2s, D_2, tid);
    for (int i = tid; i < DIM * 80; i += 256) {
        int k = i / 80, col = i - 80 * k;
        int t = col / 5, mm = col - 5 * t;
        Bq[k * 80 + col] = quants[(size_t)(a0 + t) * (9 * DIM) + (4 + mm) * DIM + k];
    }
    wait_async0();
    __syncthreads();

    // d part: 5 column tiles
    for (int ct = 0; ct < 5; ++ct) {
        v8f acc1 = {}, acc2 = {};
        acc1 = wmma_kcol(&M1s[(wave * 16) * LDP], LDP, &Bq[ct * 16], 80, acc1, lane);
        acc2 = wmma_kcol(&M2s[(wave * 16) * LDP], LDP, &Bq[ct * 16], 80, acc2, lane);
        const int col = ct * 16 + mr;
        const int t = col / 5;
#pragma unroll
        for (int vr = 0; vr < 8; ++vr)
            atomicAdd(&accD[t * 129 + wave * 16 + vr + 8 * hi], acc1[vr] * acc2[vr]);
    }
    __syncthreads();

    for (int i = tid; i < 16 * DIM; i += 256) {
        int t = i >> 7, d = i & 127;
        inp[(size_t)(a0 + t) * DIM + d] = accD[t * 129 + d];
    }
}

// ---------------------------------------------------------------------------
extern "C" void kernel_launch(void* const* d_in, const int* in_sizes, int n_in,
                              void* d_out, int out_size, void* d_ws, size_t ws_size,
                              hipStream_t stream) {
    const float* xyz     = (const float*)d_in[0];
    const float* x_tilde = (const float*)d_in[1];
    const int*   nbrs    = (const int*)d_in[2];
    const float* W1      = (const float*)d_in[3];
    const float* b1      = (const float*)d_in[4];
    const float* W2      = (const float*)d_in[5];
    const float* b2      = (const float*)d_in[6];
    const float* Wout    = (const float*)d_in[7];
    const float* act_a   = (const float*)d_in[8];
    const float* act_b   = (const float*)d_in[9];
    const float* G_s     = (const float*)d_in[10];
    const float* G_p     = (const float*)d_in[11];
    const float* G_d     = (const float*)d_in[12];
    const float* P_1     = (const float*)d_in[13];
    const float* P_2     = (const float*)d_in[14];
    const float* D_1     = (const float*)d_in[15];
    const float* D_2     = (const float*)d_in[16];
    const float* gamma   = (const float*)d_in[17];

    const int N = NATOMS;
    const size_t ND = (size_t)N * DIM;
    float* ws = (float*)d_ws;
    float* feat0  = ws;
    float* feat1  = feat0 + ND;
    float* feat2  = feat1 + ND;
    float* feat3  = feat2 + ND;
    float* quants = feat3 + ND;                 // N x 9 x 128
    float* inp    = quants + (size_t)N * 9 * DIM;

    hipMemsetAsync(quants, 0, (size_t)N * 9 * DIM * sizeof(float), stream);

    dim3 blk(256);
    dim3 grd(N / 64);
    float* feats[4] = {feat0, feat1, feat2, feat3};
    // c_term (set 0) + the three edge-feature MLPs hoisted to atoms (sets 1..3):
    // mlp(x_tilde[dst]) == mlp(x_tilde)[dst]
    for (int s = 0; s < 4; ++s) {
        resmlp_kernel<<<grd, blk, 0, stream>>>(
            x_tilde,
            W1 + (size_t)s * DIM * DIM, b1 + s * DIM,
            W2 + (size_t)s * DIM * DIM, b2 + s * DIM,
            Wout + (size_t)s * DIM * DIM,
            act_a + s * 3, act_b + s * 3, feats[s]);
    }

    edge_kernel<<<dim3(2048), blk, 0, stream>>>(xyz, nbrs, feat1, feat2, feat3,
                                                G_s, G_p, G_d, gamma, quants, NEDGES);

    inner_kernel<<<dim3(N / 16), blk, 0, stream>>>(feat0, quants, P_1, P_2, D_1, D_2, inp);

    resmlp_kernel<<<grd, blk, 0, stream>>>(
        inp,
        W1 + (size_t)4 * DIM * DIM, b1 + 4 * DIM,
        W2 + (size_t)4 * DIM * DIM, b2 + 4 * DIM,
        Wout + (size_t)4 * DIM * DIM,
        act_a + 4 * 3, act_b + 4 * 3, (float*)d_out);
}